// torch_GRL_Distributional_2465311228174
// MI455X (gfx1250) — compile-verified
//
#include <hip/hip_runtime.h>
#include <stdint.h>

// ---------------------------------------------------------------------------
// CDNA5 / gfx1250 (MI455X): wave32, WMMA 16x16x4 f32 matrix pipe.
//
// Roofline: the 16384x16384 f32 adjacency is 1 GiB. The reference needs it
// only as (adj != 0), so pass 1 (degree) streams the 1 GiB ONCE and emits a
// 32 MB bitmask via wave32 ballot; the SpMM pass (A_hat @ Y, 17.2 GFLOP f32)
// then reads 32 MB of bits + an LDS-staged Y tile shared by all 8 waves of a
// block. Total HBM traffic ~1.07 GiB (~46 us @ 23.3 TB/s) vs 2 GiB naive.
// All GEMMs run on v_wmma_f32_16x16x4_f32 — full f32 precision; lower
// precision WMMA cannot reduce the dominant (f32 input) adjacency bytes.
// ---------------------------------------------------------------------------

typedef __attribute__((ext_vector_type(2))) float v2f;
typedef __attribute__((ext_vector_type(8))) float v8f;

#define NROWS 16384
#define NACT  8
#define NATOMS 51
#define OUTW  (NACT * NATOMS)   // 408
#define KSPLIT 4                // K-dim split of the SpMM (blockIdx.y)
#define KCH   128               // staged K-chunk (columns) per block iteration
#define WPR   (NROWS / 32)      // bitmask words per row = 512

__device__ __forceinline__ v8f wmma_f32_16x16x4(v2f a, v2f b, v8f c) {
  // 8 args: (neg_a, A, neg_b, B, c_mod, C, reuse_a, reuse_b)
  return __builtin_amdgcn_wmma_f32_16x16x4_f32(false, a, false, b, (short)0, c,
                                               false, false);
}

// ---------------------------------------------------------------------------
// Pass 1 (the only full 1 GiB read): per-row nnz count -> dinv = rsqrt(deg),
// plus binarized adjacency packed to bits with wave32 ballot.
// Bit layout per row, per 128-col group g (4 words): word W_j (j=0..3),
// bit l  <->  col g*128 + 4*l + j. This makes the degree pass float4-
// coalesced AND the SpMM unpack a single shift per WMMA A element.
// ---------------------------------------------------------------------------
__global__ void degree_bits_kernel(const float* __restrict__ adj,
                                   float* __restrict__ dinv,
                                   uint32_t* __restrict__ bits, int n) {
  __shared__ float red[256];
  const int row = blockIdx.x;
  const int t = threadIdx.x;
  const int lane = t & 31;
  const int wave = t >> 5;
  const float* r = adj + (size_t)row * n;
  uint32_t* brow = bits + (size_t)row * WPR;
  float cnt = 0.f;
  for (int g = wave; g < n / 128; g += 8) {
    const float* p = r + g * 128 + lane * 4;
    __builtin_prefetch(p + 4096, 0, 0);           // stream-ahead prefetch
    const float4 v = *(const float4*)p;
    const bool p0 = (v.x != 0.f), p1 = (v.y != 0.f);
    const bool p2 = (v.z != 0.f), p3 = (v.w != 0.f);
    const uint32_t w0 = __builtin_amdgcn_ballot_w32(p0);
    const uint32_t w1 = __builtin_amdgcn_ballot_w32(p1);
    const uint32_t w2 = __builtin_amdgcn_ballot_w32(p2);
    const uint32_t w3 = __builtin_amdgcn_ballot_w32(p3);
    if (lane == 0) *(uint4*)(brow + g * 4) = make_uint4(w0, w1, w2, w3);
    cnt += (p0 ? 1.f : 0.f) + (p1 ? 1.f : 0.f) + (p2 ? 1.f : 0.f) +
           (p3 ? 1.f : 0.f);
  }
  red[t] = cnt;
  __syncthreads();
  for (int s = 128; s > 0; s >>= 1) {
    if (t < s) red[t] += red[t + s];
    __syncthreads();
  }
  if (t == 0) {
    float deg = red[0];
    if (r[row] == 0.f) deg += 1.f;  // gcn_norm forces the self-loop
    dinv[row] = rsqrtf(deg);
  }
}

// ---------------------------------------------------------------------------
// Generic dense layer on the WMMA pipe:  out = act((A[nr,K] @ W[K,NOUT]) + b)
// optionally scaled per-row (dinv / mask). 8 waves/block, one 16-row tile
// per wave.
//   A layout: lane l holds (m = l&15, k = v + 2*(l>>4))   v = 0,1
//   B layout: lane l holds (n = l&15, k = v + 2*(l>>4))
//   C/D:      reg i, lane l -> (m = i + 8*(l>>4), n = l&15)
// ---------------------------------------------------------------------------
template <int K, int NOUT, bool RELU>
__global__ void dense_wmma_kernel(const float* __restrict__ A,
                                  const float* __restrict__ W,
                                  const float* __restrict__ bias,
                                  const float* __restrict__ rowscale,
                                  float* __restrict__ out) {
  const int lane = threadIdx.x & 31;
  const int wave = threadIdx.x >> 5;
  const int row0 = (blockIdx.x * 8 + wave) * 16;
  const int lo   = lane & 15;
  const int khi  = (lane >> 4) * 2;
  const int NT = (NOUT + 15) / 16;

  for (int nt = 0; nt < NT; ++nt) {
    const int col = nt * 16 + lo;
    const bool colok = (col < NOUT);
    v8f acc = {};
#pragma unroll 8
    for (int k0 = 0; k0 < K; k0 += 4) {
      v2f a, b;
      const float* ap = A + (size_t)(row0 + lo) * K + k0 + khi;
      a.x = ap[0];
      a.y = ap[1];
      b.x = colok ? W[(size_t)(k0 + khi + 0) * NOUT + col] : 0.f;
      b.y = colok ? W[(size_t)(k0 + khi + 1) * NOUT + col] : 0.f;
      acc = wmma_f32_16x16x4(a, b, acc);
    }
    const int mbase = row0 + (lane >> 4) * 8;
    const float bv = (bias && colok) ? bias[col] : 0.f;
#pragma unroll
    for (int i = 0; i < 8; ++i) {
      float v = acc[i] + bv;
      if (rowscale) v *= rowscale[mbase + i];
      if (RELU) v = fmaxf(v, 0.f);
      if (colok) out[(size_t)(mbase + i) * NOUT + col] = v;
    }
  }
}

// ---------------------------------------------------------------------------
// SpMM over the bitmask: Spart[ky] = A_bin[:, kchunk] @ Y[kchunk, :].
// grid (NROWS/128, KSPLIT); 8 waves/block, each wave owns a 16-row tile;
// all waves walk the SAME K range so the Y tile (KCH x 32, padded stride 40,
// bank-conflict free for the half-wave-split B reads) is staged once in LDS
// and reused 8x. Self-loop handling is deferred to the combine kernel.
// ---------------------------------------------------------------------------
__global__ void gcn_spmm_bits_kernel(const uint32_t* __restrict__ bits,
                                     const float* __restrict__ Y,  // [N,32]
                                     float* __restrict__ Spart) {  // [KSPLIT,N,32]
  __shared__ float ytile[KCH * 40];  // 20 KB
  const int lane = threadIdx.x & 31;
  const int wave = threadIdx.x >> 5;
  const int r0 = (blockIdx.x * 8 + wave) * 16;
  const int mrow = lane & 15;
  const int khi = (lane >> 4) * 2;
  const int row = r0 + mrow;
  const int kbeg = blockIdx.y * (NROWS / KSPLIT);
  const int kend = kbeg + (NROWS / KSPLIT);
  const uint32_t* wrow = bits + (size_t)row * WPR;

  v8f acc0 = {}, acc1 = {};
  for (int cb = kbeg; cb < kend; cb += KCH) {
    __syncthreads();  // previous chunk's LDS reads done
    // cooperative stage: Y[cb .. cb+KCH) x 32 -> LDS (float4 coalesced)
    for (int i = threadIdx.x; i < KCH * 8; i += 256) {
      const int k = i >> 3;
      const int n4 = (i & 7) * 4;
      const float4 v = *(const float4*)(Y + (size_t)(cb + k) * 32 + n4);
      float* dst = &ytile[k * 40 + n4];
      dst[0] = v.x; dst[1] = v.y; dst[2] = v.z; dst[3] = v.w;
    }
    __syncthreads();
    // 4 bit-words cover this lane's row for the whole 128-col chunk
    const uint4 wv = *(const uint4*)(wrow + (cb >> 5));
    const uint32_t w[4] = {wv.x, wv.y, wv.z, wv.w};
#pragma unroll 8
    for (int s = 0; s < KCH / 4; ++s) {  // K=4 per WMMA
      // A element (m=mrow, k=khi+v) at col cb+4s+khi+v -> word khi+v, bit s
      v2f a;
      a.x = (float)((w[khi + 0] >> s) & 1u);
      a.y = (float)((w[khi + 1] >> s) & 1u);
      const float* yb = &ytile[(s * 4 + khi) * 40];
      v2f b0, b1;
      b0.x = yb[mrow];       b0.y = yb[40 + mrow];
      b1.x = yb[16 + mrow];  b1.y = yb[56 + mrow];
      acc0 = wmma_f32_16x16x4(a, b0, acc0);
      acc1 = wmma_f32_16x16x4(a, b1, acc1);
    }
  }
  // store partial tile (C/D layout: reg i -> m = i + 8*(lane>>4), n = lane&15)
  float* sp = Spart + (size_t)blockIdx.y * NROWS * 32;
  const int mbase = r0 + (lane >> 4) * 8;
#pragma unroll
  for (int i = 0; i < 8; ++i) {
    sp[(size_t)(mbase + i) * 32 + (lane & 15)] = acc0[i];
    sp[(size_t)(mbase + i) * 32 + 16 + (lane & 15)] = acc1[i];
  }
}

// combine K-split partials + deferred self-loop, then Xg = relu(dinv*S + bg)
__global__ void gcn_combine_kernel(const float* __restrict__ Spart,
                                   const uint32_t* __restrict__ bits,
                                   const float* __restrict__ Y,
                                   const float* __restrict__ dinv,
                                   const float* __restrict__ bg,
                                   float* __restrict__ Xg) {
  const int idx = blockIdx.x * blockDim.x + threadIdx.x;  // row*32 + n
  const int row = idx >> 5;
  const int n = idx & 31;
  float s = 0.f;
#pragma unroll
  for (int p = 0; p < KSPLIT; ++p) s += Spart[(size_t)p * NROWS * 32 + idx];
  // diag col=row -> group row>>7, word row&3, bit (row>>2)&31
  const uint32_t dw = bits[(size_t)row * WPR + ((row >> 7) << 2) + (row & 3)];
  if (((dw >> ((row >> 2) & 31)) & 1u) == 0u) s += Y[idx];  // forced self-loop
  const float v = dinv[row] * s + bg[n];
  Xg[idx] = fmaxf(v, 0.f);
}

// Fc = concat([Xd, X], axis=1) -> [N, 64]
__global__ void concat_kernel(const float* __restrict__ Xd,
                              const float* __restrict__ X,
                              float* __restrict__ Fc, int n) {
  int idx = blockIdx.x * blockDim.x + threadIdx.x;
  if (idx < n * 64) {
    int row = idx >> 6, c = idx & 63;
    Fc[idx] = (c < 32) ? Xd[row * 32 + c] : X[row * 32 + (c - 32)];
  }
}

// per (row, action): softmax over 51 atoms, clamp 0.001, expectation
__global__ void dist_head_kernel(const float* __restrict__ logits,
                                 float* __restrict__ out) {
  const int idx = blockIdx.x * blockDim.x + threadIdx.x;  // row*NACT + act
  const float* l = logits + (size_t)(idx >> 3) * OUTW + (idx & 7) * NATOMS;
  float m = l[0];
#pragma unroll
  for (int i = 1; i < NATOMS; ++i) m = fmaxf(m, l[i]);
  float s = 0.f;
#pragma unroll
  for (int i = 0; i < NATOMS; ++i) s += __expf(l[i] - m);
  const float inv = 1.f / s;
  float acc = 0.f;
#pragma unroll
  for (int i = 0; i < NATOMS; ++i) {
    float q = fmaxf(__expf(l[i] - m) * inv, 0.001f);
    acc += q * (-10.f + 0.4f * (float)i);  // support = linspace(-10,10,51)
  }
  out[idx] = acc;
}

// ---------------------------------------------------------------------------
extern "C" void kernel_launch(void* const* d_in, const int* in_sizes, int n_in,
                              void* d_out, int out_size, void* d_ws,
                              size_t ws_size, hipStream_t stream) {
  const float* features = (const float*)d_in[0];
  const float* adjacency = (const float*)d_in[1];
  const float* mask = (const float*)d_in[2];
  const float* W1 = (const float*)d_in[3];
  const float* b1 = (const float*)d_in[4];
  const float* W2 = (const float*)d_in[5];
  const float* b2 = (const float*)d_in[6];
  const float* Wg = (const float*)d_in[7];
  const float* bg = (const float*)d_in[8];
  const float* Wd = (const float*)d_in[9];
  const float* bd = (const float*)d_in[10];
  const float* Wp1 = (const float*)d_in[11];
  const float* bp1 = (const float*)d_in[12];
  const float* Wp2 = (const float*)d_in[13];
  const float* bp2 = (const float*)d_in[14];
  const float* Wo = (const float*)d_in[15];
  const float* bo = (const float*)d_in[16];
  float* out = (float*)d_out;

  // workspace partition (floats), with lifetime-based aliasing:
  //   bits region is reused for the logits; Spart region is reused for Fc.
  float* ws = (float*)d_ws;
  const size_t N = NROWS;
  float* dinv = ws;                 ws += N;
  float* X1   = ws;                 ws += N * 32;  // later reused as P
  float* X    = ws;                 ws += N * 32;
  float* Y    = ws;                 ws += N * 32;
  float* Xg   = ws;                 ws += N * 32;  // later reused as P2
  float* Xd   = ws;                 ws += N * 32;
  uint32_t* bits = (uint32_t*)ws;   ws += N * WPR;    // 32 MB; later = Lg
  float* Spart = ws;                ws += (size_t)KSPLIT * N * 32;  // later = Fc
  float* P  = X1;
  float* P2 = Xg;
  float* Lg = (float*)bits;   // N*408 <= N*512
  float* Fc = Spart;          // N*64  <= N*128

  const dim3 dblk(256);
  const dim3 dgrid(NROWS / 128);

  // 1) the single full 1 GiB pass: degrees + bitmask
  degree_bits_kernel<<<NROWS, dblk, 0, stream>>>(adjacency, dinv, bits, NROWS);
  // 2) encoder
  dense_wmma_kernel<128, 32, true><<<dgrid, dblk, 0, stream>>>(
      features, W1, b1, nullptr, X1);
  dense_wmma_kernel<32, 32, true><<<dgrid, dblk, 0, stream>>>(
      X1, W2, b2, nullptr, X);
  // 3) Y = dinv * (X @ Wg)
  dense_wmma_kernel<32, 32, false><<<dgrid, dblk, 0, stream>>>(
      X, Wg, nullptr, dinv, Y);
  // 4) S = A_bin @ Y over the 32 MB bitmask, K split 4-way (deterministic)
  gcn_spmm_bits_kernel<<<dim3(NROWS / 128, KSPLIT), dblk, 0, stream>>>(
      bits, Y, Spart);
  gcn_combine_kernel<<<(NROWS * 32) / 256, dblk, 0, stream>>>(
      Spart, bits, Y, dinv, bg, Xg);
  // 5) Xd = relu(Xg @ Wd + bd)
  dense_wmma_kernel<32, 32, true><<<dgrid, dblk, 0, stream>>>(
      Xg, Wd, bd, nullptr, Xd);
  // 6) Fc = [Xd, X]   (Spart region is free now)
  concat_kernel<<<(NROWS * 64 + 255) / 256, dblk, 0, stream>>>(Xd, X, Fc,
                                                               NROWS);
  // 7) policy MLP (X1 region free -> P; Xg region free -> P2)
  dense_wmma_kernel<64, 32, true><<<dgrid, dblk, 0, stream>>>(
      Fc, Wp1, bp1, nullptr, P);
  dense_wmma_kernel<32, 32, true><<<dgrid, dblk, 0, stream>>>(
      P, Wp2, bp2, nullptr, P2);
  // 8) masked logits = (P2 @ Wo + bo) * mask[row]  (bits region free -> Lg)
  dense_wmma_kernel<32, OUTW, false><<<dgrid, dblk, 0, stream>>>(
      P2, Wo, bo, mask, Lg);
  // 9) distributional head -> [N, NACT]
  dist_head_kernel<<<(NROWS * NACT) / 256, dblk, 0, stream>>>(Lg, out);

  (void)in_sizes; (void)n_in; (void)out_size; (void)ws_size;
}